// WeightedHausdorffDistance_9809705304668
// MI455X (gfx1250) — compile-verified
//
#include <hip/hip_runtime.h>
#include <math.h>

// Problem constants (match reference)
#define HH 256
#define WW 256
#define BB 8
#define GG 128
#define NN (HH * WW)
#define NSTRIPS (NN / 16)          // 4096 strips of 16 pixels per batch item
#define MAXD 362.03867196751236f   // sqrt(256^2 + 256^2)

#define BLOCKS_PER_B 64
#define THREADS 256                // 8 waves (wave32)
#define WAVES (THREADS / 32)

typedef float v2f __attribute__((ext_vector_type(2)));
typedef float v8f __attribute__((ext_vector_type(8)));

__global__ void whd_init(float* __restrict__ w, int n) {
    int i = blockIdx.x * blockDim.x + threadIdx.x;
    if (i < n) w[i] = 0.0f;
}

__global__ __launch_bounds__(THREADS)
void whd_main(const float* __restrict__ pm,   // [B,H,W]
              const float* __restrict__ gt,   // [B,G,2]
              const float* __restrict__ osz,  // [B,2]
              float* __restrict__ S,          // [B,G] sum over pixels of (wd+1e-6)^-9
              float* __restrict__ T1,         // [B]   sum p * min_d
              float* __restrict__ NE)         // [B]   sum p
{
    __shared__ float sSum[GG];
    __shared__ float sT1;
    __shared__ float sNE;

    const int b = blockIdx.y;
    const int lane = threadIdx.x & 31;
    const int waveInBlk = threadIdx.x >> 5;
    const int gwave = blockIdx.x * WAVES + waveInBlk;
    const int waveStride = BLOCKS_PER_B * WAVES;

    for (int i = threadIdx.x; i < GG; i += THREADS) sSum[i] = 0.0f;
    if (threadIdx.x == 0) { sT1 = 0.0f; sNE = 0.0f; }
    __syncthreads();

    const float nr = osz[b * 2 + 0] * (1.0f / (float)HH);
    const float nc = osz[b * 2 + 1] * (1.0f / (float)WW);
    const int m = lane & 15;
    const bool hi = lane >= 16;

    // B-matrices for all 8 g-tiles, resident in registers for the whole kernel.
    // f32 4x16 B layout (mirrors documented A layout): VGPR0 = {K0 | K2}, VGPR1 = {K1 | K3}
    // K rows: [-2*yr, -2*yc, 1, yr^2+yc^2]
    v2f Bt[8];
#pragma unroll
    for (int j = 0; j < 8; ++j) {
        int g = j * 16 + m;
        float yr = gt[(b * GG + g) * 2 + 0] * nr;
        float yc = gt[(b * GG + g) * 2 + 1] * nc;
        Bt[j].x = hi ? 1.0f : (-2.0f * yr);
        Bt[j].y = hi ? (yr * yr + yc * yc) : (-2.0f * yc);
    }

    float accg[8];
#pragma unroll
    for (int j = 0; j < 8; ++j) accg[j] = 0.0f;
    float acc_t1 = 0.0f, acc_ne = 0.0f;

    const float* pmb = pm + b * NN;

    for (int strip = gwave; strip < NSTRIPS; strip += waveStride) {
        const int base = strip * 16;
        const int row = base >> 8;     // 16 | 256, so one strip = 16 cols of one row
        const int col0 = base & 255;
        const float xr = (float)row * nr;
        const float xc = (float)(col0 + m) * nc;
        // A 16x4 layout: VGPR0 = {K0: xr | K2: xr^2+xc^2}, VGPR1 = {K1: xc | K3: 1}
        v2f A;
        A.x = hi ? (xr * xr + xc * xc) : xr;
        A.y = hi ? 1.0f : xc;

        // 8 p-values for this lane's pixel half (M = v + 8*hi), 16B-aligned float4 loads
        float pv[8];
        const float* pp = pmb + base + (hi ? 8 : 0);
        float4 p0 = *(const float4*)(pp);
        float4 p1 = *(const float4*)(pp + 4);
        pv[0] = p0.x; pv[1] = p0.y; pv[2] = p0.z; pv[3] = p0.w;
        pv[4] = p1.x; pv[5] = p1.y; pv[6] = p1.z; pv[7] = p1.w;

        float minv[8];
#pragma unroll
        for (int v = 0; v < 8; ++v) minv[v] = 3.0e38f;

#pragma unroll
        for (int j = 0; j < 8; ++j) {
            v8f c = {};
            // d^2 tile: 16 pixels x 16 GT points via |x|^2+|y|^2-2x.y, K=4
            v8f d = __builtin_amdgcn_wmma_f32_16x16x4_f32(
                false, A, false, Bt[j], (short)0, c, false, false);
            float a = 0.0f;
#pragma unroll
            for (int v = 0; v < 8; ++v) {
                float dsq = d[v];
                minv[v] = fminf(minv[v], dsq);
                float dd = sqrtf(fmaxf(dsq, 0.0f));
                // wd = (1-p)*MAXD + p*d + 1e-6 = MAXD + p*(d - MAXD) + 1e-6
                float wd = fmaf(pv[v], dd - MAXD, MAXD) + 1e-6f;
                float t = 1.0f / wd;
                float t2 = t * t, t4 = t2 * t2, t8 = t4 * t4;
                a += t8 * t;               // (wd+1e-6)^-9
            }
            accg[j] += a;
        }

        // term1: min over all 128 g = min over lanes within each 16-lane half
#pragma unroll
        for (int v = 0; v < 8; ++v) {
            float x = minv[v];
            x = fminf(x, __shfl_xor(x, 1, 16));
            x = fminf(x, __shfl_xor(x, 2, 16));
            x = fminf(x, __shfl_xor(x, 4, 16));
            x = fminf(x, __shfl_xor(x, 8, 16));
            minv[v] = x;
        }
        float c1 = 0.0f, ne = 0.0f;
#pragma unroll
        for (int v = 0; v < 8; ++v) {
            c1 += pv[v] * sqrtf(fmaxf(minv[v], 0.0f));
            ne += pv[v];
        }
        if (m == 0) { acc_t1 += c1; acc_ne += ne; }  // lanes 0 & 16: pixel halves
    }

    // Block-level reduction: per-g bins via LDS float atomics, then one global atomic/bin
#pragma unroll
    for (int j = 0; j < 8; ++j)
        atomicAdd(&sSum[j * 16 + m], accg[j]);
    if (m == 0) { atomicAdd(&sT1, acc_t1); atomicAdd(&sNE, acc_ne); }
    __syncthreads();

    if (threadIdx.x < GG) atomicAdd(&S[b * GG + threadIdx.x], sSum[threadIdx.x]);
    if (threadIdx.x == 0) { atomicAdd(&T1[b], sT1); atomicAdd(&NE[b], sNE); }
}

__global__ __launch_bounds__(128)
void whd_final(const float* __restrict__ S, const float* __restrict__ T1,
               const float* __restrict__ NE, float* __restrict__ out)
{
    __shared__ float red[GG];
    const int g = threadIdx.x;  // 0..127
    float t2 = 0.0f;
    for (int b = 0; b < BB; ++b) {
        float mean = S[b * GG + g] * (1.0f / (float)NN);
        red[g] = powf(mean, -1.0f / 9.0f);
        __syncthreads();
        for (int s = 64; s > 0; s >>= 1) {
            if (g < s) red[g] += red[g + s];
            __syncthreads();
        }
        if (g == 0) t2 += red[0] * (1.0f / (float)GG);
        __syncthreads();
    }
    if (g == 0) {
        float t1 = 0.0f;
        for (int b = 0; b < BB; ++b) t1 += T1[b] / (NE[b] + 1e-6f);
        out[0] = t1 * (1.0f / (float)BB) + t2 * (1.0f / (float)BB);
    }
}

extern "C" void kernel_launch(void* const* d_in, const int* in_sizes, int n_in,
                              void* d_out, int out_size, void* d_ws, size_t ws_size,
                              hipStream_t stream) {
    const float* pm  = (const float*)d_in[0];  // [B,H,W]
    const float* gt  = (const float*)d_in[1];  // [B,G,2]
    const float* osz = (const float*)d_in[2];  // [B,2]
    float* w  = (float*)d_ws;
    float* S  = w;                  // B*G
    float* T1 = w + BB * GG;        // B
    float* NE = T1 + BB;            // B
    float* out = (float*)d_out;

    const int nws = BB * GG + 2 * BB;
    whd_init<<<(nws + 255) / 256, 256, 0, stream>>>(w, nws);

    dim3 grid(BLOCKS_PER_B, BB);
    whd_main<<<grid, THREADS, 0, stream>>>(pm, gt, osz, S, T1, NE);

    whd_final<<<1, 128, 0, stream>>>(S, T1, NE, out);
}